// GlottalFlowTable_61881888800992
// MI455X (gfx1250) — compile-verified
//
#include <hip/hip_runtime.h>
#include <cstdint>

// Problem constants (match the reference; hop_size input is a fixed 256)
#define BATCH    16
#define SEQ      384000
#define HOP      256
#define NHOPS    (SEQ / HOP)        // 1500
#define NFRAMES  (NHOPS + 1)        // 1501
#define NTABLES  100
#define TL       1024

// 16-byte int vector type matching the builtin's parameter type
typedef int v4i __attribute__((vector_size(16)));
typedef __attribute__((address_space(1))) v4i* gv4i_p;   // global
typedef __attribute__((address_space(3))) v4i* lv4i_p;   // LDS

// ---------------------------------------------------------------------------
// CDNA5 async global->LDS copy (16 bytes per lane).  The clang builtin exists
// on this toolchain (probe-confirmed last round); its pointer params are
// v4i32* in AS1/AS3.  Generic LDS pointers carry the LDS byte offset in their
// low 32 bits (aperture rule), so the uintptr
// trick materializes the AS-qualified pointers without an addrspacecast.
// ---------------------------------------------------------------------------
__device__ __forceinline__ void async_copy16(const float* g, float* l)
{
#if defined(__gfx1250__) && __has_builtin(__builtin_amdgcn_global_load_async_to_lds_b128)
    __builtin_amdgcn_global_load_async_to_lds_b128(
        (gv4i_p)(uintptr_t)g,
        (lv4i_p)(uint32_t)(uintptr_t)l,
        0, 0);
#elif defined(__gfx1250__)
    asm volatile("global_load_async_to_lds_b128 %0, %1, off"
                 :
                 : "v"((uint32_t)(uintptr_t)l),
                   "v"((unsigned long long)(uintptr_t)g)
                 : "memory");
#else
    *(float4*)l = *(const float4*)g;   // host-pass placeholder
#endif
}

__device__ __forceinline__ void wait_async0()
{
#if defined(__gfx1250__) && __has_builtin(__builtin_amdgcn_s_wait_asynccnt)
    __builtin_amdgcn_s_wait_asynccnt(0);
#elif defined(__gfx1250__)
    asm volatile("s_wait_asynccnt 0" ::: "memory");
#endif
}

// ---------------------------------------------------------------------------
// One 256-thread workgroup per (batch b, hop h).  Threads t = sample-in-hop.
// ---------------------------------------------------------------------------
__global__ __launch_bounds__(HOP)
void glottal_flow_table_kernel(const float* __restrict__ phase,   // (B, SEQ)
                               const float* __restrict__ tsw,     // (B, NFRAMES)
                               const float* __restrict__ table,   // (NTABLES, TL)
                               float*       __restrict__ out)     // (B, SEQ)
{
    __shared__ __align__(16) float  s_raw[4][TL];   // 16 KB: raw table rows
    __shared__ float2               s_pair[TL + 1]; // interleaved (floor, ceil)

    const int h = blockIdx.x;
    const int b = blockIdx.y;
    const int t = threadIdx.x;

    // ---- uniform (per-block) frame-select weights: frames h and h+1 -------
    const float w0   = tsw[b * NFRAMES + h];
    const float w1   = tsw[b * NFRAMES + h + 1];
    float rawf = w0 * (float)(NTABLES - 1);
    int   fif  = (int)rawf;
    fif = fif < 0 ? 0 : (fif > NTABLES - 2 ? NTABLES - 2 : fif);
    const float p1f = rawf - (float)fif;
    float rawc = w1 * (float)(NTABLES - 1);
    int   fic  = (int)rawc;
    fic = fic < 0 ? 0 : (fic > NTABLES - 2 ? NTABLES - 2 : fic);
    const float p1c = rawc - (float)fic;

    // ---- stage 1: async-DMA the four raw table rows into LDS --------------
    {
        const int e = 4 * t;                       // 16B chunk per lane
        async_copy16(table + (size_t)fif       * TL + e, &s_raw[0][e]);
        async_copy16(table + (size_t)(fif + 1) * TL + e, &s_raw[1][e]);
        async_copy16(table + (size_t)fic       * TL + e, &s_raw[2][e]);
        async_copy16(table + (size_t)(fic + 1) * TL + e, &s_raw[3][e]);
    }

    // per-thread phase load overlaps with the DMA
    const int   base = b * SEQ + h * HOP;
    const float ph   = phase[base + t];

    wait_async0();        // own wave's async writes complete
    __syncthreads();      // all waves' writes visible

    // ---- stage 2: wavetable lerp -> interleaved lookup (floor, ceil) ------
#pragma unroll
    for (int k = 0; k < TL / HOP; ++k) {
        const int i = t + k * HOP;
        const float f = s_raw[0][i] * (1.0f - p1f) + s_raw[1][i] * p1f;
        const float c = s_raw[2][i] * (1.0f - p1c) + s_raw[3][i] * p1c;
        s_pair[i] = make_float2(f, c);
    }
    if (t == 0) s_pair[TL] = s_pair[0];   // wrap sample = element 0 (concat)
    __syncthreads();

    // ---- stage 3: per-sample phase lookup + lerp + hop crossfade ----------
    const float raw2 = ph * (float)TL;
    int i0 = (int)raw2;
    i0 = i0 < 0 ? 0 : (i0 > TL - 1 ? TL - 1 : i0);
    const float p  = raw2 - (float)i0;

    const float2 a  = s_pair[i0];          // ds_load_b64
    const float2 bq = s_pair[i0 + 1];      // ds_load_b64

    const float sf = a.x  * (1.0f - p) + bq.x * p;   // floor-frame sample
    const float sc = a.y  * (1.0f - p) + bq.y * p;   // ceil-frame sample
    const float p2 = (float)t * (1.0f / (float)HOP); // exact: /256
    out[base + t] = sf * (1.0f - p2) + sc * p2;
}

// ---------------------------------------------------------------------------
extern "C" void kernel_launch(void* const* d_in, const int* in_sizes, int n_in,
                              void* d_out, int out_size, void* d_ws, size_t ws_size,
                              hipStream_t stream)
{
    const float* phase = (const float*)d_in[0];   // (16, 384000) f32
    const float* tsw   = (const float*)d_in[1];   // (16, 1501)   f32
    const float* table = (const float*)d_in[2];   // (100, 1024)  f32
    // d_in[3] = hop_size (=256), compile-time constant here
    float* out = (float*)d_out;

    dim3 grid(NHOPS, BATCH);   // 1500 x 16 = 24000 blocks
    glottal_flow_table_kernel<<<grid, HOP, 0, stream>>>(phase, tsw, table, out);

    (void)in_sizes; (void)n_in; (void)out_size; (void)d_ws; (void)ws_size;
}